// DeformableConv2d_28054726378226
// MI455X (gfx1250) — compile-verified
//
#include <hip/hip_runtime.h>

typedef __attribute__((ext_vector_type(2))) float v2f;
typedef __attribute__((ext_vector_type(8))) float v8f;

#define BATCH 8
#define CIN   64
#define COUT  64
#define HH    96
#define WW    96
#define HWSZ  (HH*WW)          // 9216
#define K2    9
#define OCOFF 18               // 2*K*K offset channels

// workspace layout (in floats)
#define XT_OFF   0
#define XT_SZ    (BATCH*HWSZ*CIN)      // 4,718,592  x in NHWC
#define OFFS_OFF (XT_OFF + XT_SZ)
#define OFFS_SZ  (BATCH*HWSZ*OCOFF)    // 1,327,104  offsets NHWC(18)
#define W2_OFF   (OFFS_OFF + OFFS_SZ)
#define W2_SZ    (K2*32*128)           // 36,864  W2p[k][c/2][o*2+e]
#define W1_OFF   (W2_OFF + W2_SZ)
#define W1_SZ    (K2*32*64)            // 18,432  W1p[k][c/2][o*2+e], o<32 (18 real)

__device__ __forceinline__ v8f wmma_f32(v2f a, v2f b, v8f c) {
    return __builtin_amdgcn_wmma_f32_16x16x4_f32(false, a, false, b, (short)0, c,
                                                 false, false);
}

// ---------------- kernel 1: NCHW -> NHWC transpose of x --------------------
__global__ void k_transpose(const float* __restrict__ x, float* __restrict__ xt) {
    int idx = blockIdx.x * 256 + threadIdx.x;           // coalesced over input
    if (idx >= BATCH * CIN * HWSZ) return;
    int hw = idx % HWSZ;
    int t  = idx / HWSZ;
    int c  = t % CIN;
    int b  = t / CIN;
    xt[(b * HWSZ + hw) * CIN + c] = x[idx];
}

// ---------------- kernel 2: weight re-layout (paired K-rows) ---------------
// W2p[(k*32+cp)*128 + 2*o + e] = conv_w[o][2*cp+e][k]   -> B frag = one b64 load
// W1p[(k*32+cp)*64  + 2*o + e] = offset_w[o][2*cp+e][k] (o<18, zero-pad to 32)
__global__ void k_wprep(const float* __restrict__ conv_w,
                        const float* __restrict__ offset_w,
                        float* __restrict__ W2p, float* __restrict__ W1p) {
    int tid = blockIdx.x * 256 + threadIdx.x;
    if (tid < K2 * 32 * 128) {
        int e  = tid & 1;
        int o  = (tid >> 1) & 63;
        int cp = (tid >> 7) & 31;
        int k  = tid >> 12;
        int c  = 2 * cp + e;
        W2p[tid] = conv_w[(o * CIN + c) * K2 + k];
    }
    if (tid < K2 * 32 * 64) {
        int e  = tid & 1;
        int o  = (tid >> 1) & 31;
        int cp = (tid >> 6) & 31;
        int k  = tid >> 11;
        int c  = 2 * cp + e;
        W1p[tid] = (o < OCOFF) ? offset_w[(o * CIN + c) * K2 + k] : 0.0f;
    }
}

// ---------------- kernel 3: offset conv (implicit GEMM, fp32 WMMA) ---------
// block = 256 threads = 8 waves; each wave owns 32 consecutive positions
// (96 % 32 == 0 -> never crosses a W row / batch boundary). 2 M-subtiles
// share every B fragment.
__global__ void k_offconv(const float* __restrict__ xt,
                          const float* __restrict__ W1p,
                          const float* __restrict__ offset_b,
                          float* __restrict__ offs) {
    __shared__ float Ald[8][32][68];     // 68-stride: conflict-free, 16B aligned

    const int wave = threadIdx.x >> 5;
    const int lane = threadIdx.x & 31;
    const int posBase = blockIdx.x * 256 + wave * 32;
    const int b  = posBase / HWSZ;
    const int hw = posBase % HWSZ;
    const int h  = hw / WW;
    const int w0 = hw % WW;
    const int half = lane >> 4;
    const int m    = lane & 15;

    v8f acc[2][2];                        // [Mtile][Ntile]
    {
        float b0 = offset_b[m];
        float b1 = (m < 2) ? offset_b[16 + m] : 0.0f;
        #pragma unroll
        for (int r = 0; r < 8; ++r) {
            acc[0][0][r] = b0; acc[1][0][r] = b0;
            acc[0][1][r] = b1; acc[1][1][r] = b1;
        }
    }

    for (int k = 0; k < K2; ++k) {
        const int ky = k / 3, kx = k % 3;
        const int yy = h + ky - 1;
        const bool rowv = (yy >= 0) && (yy < HH);
        // stage A chunk: 32 pos x 64 ch (float4 granularity)
        #pragma unroll
        for (int it = 0; it < 16; ++it) {
            int g  = it * 32 + lane;
            int p  = g >> 4;
            int cg = g & 15;
            int xx = w0 + p + kx - 1;
            float4 v = make_float4(0.f, 0.f, 0.f, 0.f);
            if (rowv && xx >= 0 && xx < WW)
                v = *(const float4*)(xt + ((size_t)(b * HWSZ + yy * WW + xx)) * CIN + cg * 4);
            *(float4*)&Ald[wave][p][cg * 4] = v;
        }
        __syncthreads();
        // 16 K-steps x 2 N-tiles x 2 M-tiles of v_wmma_f32_16x16x4_f32
        #pragma unroll
        for (int kk = 0; kk < 16; ++kk) {
            int c0 = kk * 4 + half * 2;
            int cp = kk * 2 + half;
            v2f a0 = *(const v2f*)&Ald[wave][m][c0];
            v2f a1 = *(const v2f*)&Ald[wave][16 + m][c0];
            const float* wr = W1p + (size_t)(k * 32 + cp) * 64;
            v2f b0 = *(const v2f*)(wr + 2 * m);
            v2f b1 = *(const v2f*)(wr + 2 * (16 + m));
            acc[0][0] = wmma_f32(a0, b0, acc[0][0]);
            acc[1][0] = wmma_f32(a1, b0, acc[1][0]);
            acc[0][1] = wmma_f32(a0, b1, acc[0][1]);
            acc[1][1] = wmma_f32(a1, b1, acc[1][1]);
        }
        __syncthreads();
    }
    // store offsets NHWC(18); D layout: vgpr r -> M = r + 8*half, N = m
    #pragma unroll
    for (int mt = 0; mt < 2; ++mt) {
        #pragma unroll
        for (int r = 0; r < 8; ++r) {
            int pos = posBase + mt * 16 + r + 8 * half;
            offs[pos * OCOFF + m] = acc[mt][0][r];
            if (m < 2) offs[pos * OCOFF + 16 + m] = acc[mt][1][r];
        }
    }
}

// ---------------- kernel 4: deformable sampling + fp32 WMMA GEMM -----------
__global__ void k_deform(const float* __restrict__ xt,
                         const float* __restrict__ offs,
                         const float* __restrict__ W2p,
                         const float* __restrict__ conv_b,
                         float* __restrict__ out) {
    __shared__ float Ald[8][32][68];   // per-wave 32 x 64 A-chunk (one tap)
    __shared__ float P[8][32][8];      // per-(pos,tap): 4 addrs + 4 bilinear weights

    const int wave = threadIdx.x >> 5;
    const int lane = threadIdx.x & 31;
    const int posBase = blockIdx.x * 256 + wave * 32;
    const int b  = posBase / HWSZ;
    const int hw = posBase % HWSZ;
    const int h  = hw / WW;
    const int w0 = hw % WW;
    const int half = lane >> 4;
    const int m    = lane & 15;

    v8f acc[2][4];                      // [Mtile][Ntile]
    #pragma unroll
    for (int n = 0; n < 4; ++n) {
        float bv = conv_b[n * 16 + m];
        #pragma unroll
        for (int r = 0; r < 8; ++r) { acc[0][n][r] = bv; acc[1][n][r] = bv; }
    }

    for (int k = 0; k < K2; ++k) {
        const int ky = k / 3, kx = k % 3;
        // ---- phase 1: per-position sampling parameters (all 32 lanes) ----
        {
            int pos = posBase + lane;
            int w   = w0 + lane;
            float dy = offs[pos * OCOFF + 2 * k];
            float dx = offs[pos * OCOFF + 2 * k + 1];
            float py = (float)(h - 1 + ky) + dy;
            float px = (float)(w - 1 + kx) + dx;
            float y0f = floorf(py), x0f = floorf(px);
            float wy = py - y0f, wx = px - x0f;
            int y0 = (int)y0f, x0 = (int)x0f;
            int y1 = y0 + 1,   x1 = x0 + 1;
            bool vy0 = (y0 >= 0) && (y0 < HH), vy1 = (y1 >= 0) && (y1 < HH);
            bool vx0 = (x0 >= 0) && (x0 < WW), vx1 = (x1 >= 0) && (x1 < WW);
            int y0c = min(max(y0, 0), HH - 1), y1c = min(max(y1, 0), HH - 1);
            int x0c = min(max(x0, 0), WW - 1), x1c = min(max(x1, 0), WW - 1);
            int base = b * HWSZ;
            P[wave][lane][0] = __int_as_float((base + y0c * WW + x0c) * CIN);
            P[wave][lane][1] = __int_as_float((base + y0c * WW + x1c) * CIN);
            P[wave][lane][2] = __int_as_float((base + y1c * WW + x0c) * CIN);
            P[wave][lane][3] = __int_as_float((base + y1c * WW + x1c) * CIN);
            P[wave][lane][4] = (vy0 && vx0) ? (1.f - wy) * (1.f - wx) : 0.f;
            P[wave][lane][5] = (vy0 && vx1) ? (1.f - wy) * wx         : 0.f;
            P[wave][lane][6] = (vy1 && vx0) ? wy * (1.f - wx)         : 0.f;
            P[wave][lane][7] = (vy1 && vx1) ? wy * wx                 : 0.f;
        }
        __syncthreads();
        // ---- phase 2: bilinear gather/blend into A chunk (float4/lane) ----
        #pragma unroll
        for (int it = 0; it < 16; ++it) {
            int g  = it * 32 + lane;
            int p  = g >> 4;
            int cg = g & 15;
            int a00 = __float_as_int(P[wave][p][0]);
            int a01 = __float_as_int(P[wave][p][1]);
            int a10 = __float_as_int(P[wave][p][2]);
            int a11 = __float_as_int(P[wave][p][3]);
            float q00 = P[wave][p][4], q01 = P[wave][p][5];
            float q10 = P[wave][p][6], q11 = P[wave][p][7];
            float4 v00 = *(const float4*)(xt + a00 + cg * 4);
            float4 v01 = *(const float4*)(xt + a01 + cg * 4);
            float4 v10 = *(const float4*)(xt + a10 + cg * 4);
            float4 v11 = *(const float4*)(xt + a11 + cg * 4);
            float4 s;
            s.x = fmaf(q00, v00.x, fmaf(q01, v01.x, fmaf(q10, v10.x, q11 * v11.x)));
            s.y = fmaf(q00, v00.y, fmaf(q01, v01.y, fmaf(q10, v10.y, q11 * v11.y)));
            s.z = fmaf(q00, v00.z, fmaf(q01, v01.z, fmaf(q10, v10.z, q11 * v11.z)));
            s.w = fmaf(q00, v00.w, fmaf(q01, v01.w, fmaf(q10, v10.w, q11 * v11.w)));
            *(float4*)&Ald[wave][p][cg * 4] = s;
        }
        __syncthreads();
        // ---- phase 3: 16 K-steps x 4 N-tiles x 2 M-tiles of fp32 WMMA ----
        #pragma unroll
        for (int kk = 0; kk < 16; ++kk) {
            int c0 = kk * 4 + half * 2;
            int cp = kk * 2 + half;
            v2f a0 = *(const v2f*)&Ald[wave][m][c0];
            v2f a1 = *(const v2f*)&Ald[wave][16 + m][c0];
            const float* wr = W2p + (size_t)(k * 32 + cp) * 128;
            #pragma unroll
            for (int n = 0; n < 4; ++n) {
                v2f bf = *(const v2f*)(wr + 2 * (n * 16 + m));
                acc[0][n] = wmma_f32(a0, bf, acc[0][n]);
                acc[1][n] = wmma_f32(a1, bf, acc[1][n]);
            }
        }
    }
    // ---- store D to NCHW output ----
    #pragma unroll
    for (int n = 0; n < 4; ++n) {
        int o = n * 16 + m;
        #pragma unroll
        for (int mt = 0; mt < 2; ++mt) {
            #pragma unroll
            for (int r = 0; r < 8; ++r) {
                int M = mt * 16 + r + 8 * half;
                out[((size_t)(b * COUT + o) * HH + h) * WW + (w0 + M)] = acc[mt][n][r];
            }
        }
    }
}

extern "C" void kernel_launch(void* const* d_in, const int* in_sizes, int n_in,
                              void* d_out, int out_size, void* d_ws, size_t ws_size,
                              hipStream_t stream) {
    const float* x        = (const float*)d_in[0];
    const float* offset_w = (const float*)d_in[1];
    const float* offset_b = (const float*)d_in[2];
    const float* conv_w   = (const float*)d_in[3];
    const float* conv_b   = (const float*)d_in[4];
    float* out = (float*)d_out;
    float* ws  = (float*)d_ws;

    float* xt   = ws + XT_OFF;
    float* offs = ws + OFFS_OFF;
    float* W2p  = ws + W2_OFF;
    float* W1p  = ws + W1_OFF;

    k_transpose<<<(BATCH * CIN * HWSZ + 255) / 256, 256, 0, stream>>>(x, xt);
    k_wprep<<<(K2 * 32 * 128 + 255) / 256, 256, 0, stream>>>(conv_w, offset_w, W2p, W1p);

    const int nblocks = (BATCH * HWSZ) / 256;   // 288
    k_offconv<<<nblocks, 256, 0, stream>>>(xt, W1p, offset_b, offs);
    k_deform<<<nblocks, 256, 0, stream>>>(xt, offs, W2p, conv_b, out);
}